// QuanvolutionalLayer_64312840290382
// MI455X (gfx1250) — compile-verified
//
#include <hip/hip_runtime.h>
#include <math.h>

typedef __attribute__((ext_vector_type(2))) float v2f;
typedef __attribute__((ext_vector_type(8))) float v8f;

#define PI_F 3.14159265358979323846f

// ---------------------------------------------------------------------------
// Kernel 1: build the 16x16 complex unitary U of the weight-dependent part of
// the circuit by simulating it on each computational basis state.
// Thread k computes column k of U. Layout in ws: Ure[16][16], Uim[16][16].
// Index convention (matches reference state [B, w0, w1, w2, w3]):
//   i = b0*8 + b1*4 + b2*2 + b3   (wire q has stride 8 >> q)
// ---------------------------------------------------------------------------
__device__ __forceinline__ void apply_rx16(float* re, float* im, int st,
                                           float cw, float sw) {
#pragma unroll
  for (int i = 0; i < 16; ++i) {
    if (i & st) continue;
    int j = i | st;
    float r0 = re[i], i0 = im[i], r1 = re[j], i1 = im[j];
    // [[cw, -i sw], [-i sw, cw]]
    re[i] = cw * r0 + sw * i1;  im[i] = cw * i0 - sw * r1;
    re[j] = sw * i0 + cw * r1;  im[j] = -sw * r0 + cw * i1;
  }
}

__device__ __forceinline__ void apply_ry16(float* re, float* im, int st,
                                           float c, float s) {
#pragma unroll
  for (int i = 0; i < 16; ++i) {
    if (i & st) continue;
    int j = i | st;
    float r0 = re[i], i0 = im[i], r1 = re[j], i1 = im[j];
    // [[c, -s], [s, c]] (real gate)
    re[i] = c * r0 - s * r1;  im[i] = c * i0 - s * i1;
    re[j] = s * r0 + c * r1;  im[j] = s * i0 + c * i1;
  }
}

__device__ __forceinline__ void apply_cnot16(float* re, float* im, int cs,
                                             int ts) {
#pragma unroll
  for (int i = 0; i < 16; ++i) {
    if ((i & cs) && !(i & ts)) {
      int j = i | ts;
      float tr = re[i]; re[i] = re[j]; re[j] = tr;
      float ti = im[i]; im[i] = im[j]; im[j] = ti;
    }
  }
}

__global__ void build_unitary_kernel(const float* __restrict__ w,
                                     float* __restrict__ Ure,
                                     float* __restrict__ Uim) {
  int k = threadIdx.x;
  if (k >= 16) return;

  float re[16], im[16];
#pragma unroll
  for (int i = 0; i < 16; ++i) { re[i] = (i == k) ? 1.0f : 0.0f; im[i] = 0.0f; }

  // accurate libm sincos here: runs once on 16 threads, weights are arbitrary
  float sw0, cw0, sw1, cw1, sw2, cw2, sw3, cw3;
  sincosf(0.5f * w[0], &sw0, &cw0);
  sincosf(0.5f * w[1], &sw1, &cw1);
  sincosf(0.5f * w[2], &sw2, &cw2);
  sincosf(0.5f * w[3], &sw3, &cw3);

  apply_rx16(re, im, 8, cw0, sw0);   // RX(w0) on wire 0
  apply_rx16(re, im, 4, cw1, sw1);   // RX(w1) on wire 1
  apply_cnot16(re, im, 2, 1);        // CNOT(2,3)
  apply_cnot16(re, im, 8, 2);        // CNOT(0,2)
  apply_cnot16(re, im, 8, 1);        // CNOT(0,3)
  apply_ry16(re, im, 8, cw2, sw2);   // RY(w2) on wire 0
  apply_ry16(re, im, 1, cw3, sw3);   // RY(w3) on wire 3

#pragma unroll
  for (int i = 0; i < 16; ++i) {
    Ure[i * 16 + k] = re[i];
    Uim[i * 16 + k] = im[i];
  }
}

// ---------------------------------------------------------------------------
// Kernel 2: one wave per 16-sample tile.
//   B-matrix = psi (16 x 16 real product-state amplitudes, f32)
//   A-matrix = U.re / U.im rows (16 x 4 chunks), K looped 4x per part
//   D = A*B via v_wmma_f32_16x16x4_f32 (full f32 precision)
// Then per-lane signed reductions of |phi|^2 give the 4 Pauli-Z expectations.
// Embedding angle is (pi/2)*x with x in [0,1) -> hardware v_sin/v_cos path
// (__sinf/__cosf) needs no range reduction and is ~1 ulp here.
// ---------------------------------------------------------------------------
__global__ __launch_bounds__(128) void quanv_wmma_kernel(
    const float* __restrict__ x, const float* __restrict__ Ure,
    const float* __restrict__ Uim, float* __restrict__ out, int ntiles) {
  const int lane = threadIdx.x & 31;
  const int wave = threadIdx.x >> 5;
  const int tile = blockIdx.x * 4 + wave;
  if (tile >= ntiles) return;          // wave-uniform: EXEC stays all-ones

  const int n  = lane & 15;            // column (sample within tile) and row M
  const int hi = lane >> 4;            // half-lane selector
  const int s  = tile * 16 + n;        // global sample index

  // --- build psi column for this sample (real product state) ---
  const float4 xv = reinterpret_cast<const float4*>(x)[s];
  const float a0 = 0.5f * PI_F * xv.x;
  const float a1 = 0.5f * PI_F * xv.y;
  const float a2 = 0.5f * PI_F * xv.z;
  const float a3 = 0.5f * PI_F * xv.w;
  const float s0 = __sinf(a0), c0 = __cosf(a0);
  const float s1 = __sinf(a1), c1 = __cosf(a1);
  const float s2 = __sinf(a2), c2 = __cosf(a2);
  const float s3 = __sinf(a3), c3 = __cosf(a3);
  float p01[4] = {c0 * c1, c0 * s1, s0 * c1, s0 * s1};
  float p23[4] = {c2 * c3, c2 * s3, s2 * c3, s2 * s3};
  float psi[16];
#pragma unroll
  for (int i = 0; i < 16; ++i) psi[i] = p01[i >> 2] * p23[i & 3];

  // --- 8x v_wmma_f32_16x16x4_f32: phi.re = Ure*psi, phi.im = Uim*psi ---
  v8f cre = {};
  v8f cim = {};
#pragma unroll
  for (int kc = 0; kc < 4; ++kc) {
    const int kbase = 4 * kc + 2 * hi; // K rows held by this lane (A & B frag)
    v2f a_re, a_im, b;
    a_re.x = Ure[n * 16 + kbase];
    a_re.y = Ure[n * 16 + kbase + 1];
    a_im.x = Uim[n * 16 + kbase];
    a_im.y = Uim[n * 16 + kbase + 1];
    b.x = psi[kbase];
    b.y = psi[kbase + 1];
    cre = __builtin_amdgcn_wmma_f32_16x16x4_f32(false, a_re, false, b,
                                                (short)0, cre, false, false);
    cim = __builtin_amdgcn_wmma_f32_16x16x4_f32(false, a_im, false, b,
                                                (short)0, cim, false, false);
  }

  // --- probabilities; D element v of this lane is row M = v + 8*hi, col n ---
  float p[8];
#pragma unroll
  for (int v = 0; v < 8; ++v) p[v] = cre[v] * cre[v] + cim[v] * cim[v];

  // signed partial sums: sign_q(i) = (bit (3-q) of i == 0) ? +1 : -1
  float etot = ((p[0] + p[1]) + (p[2] + p[3])) + ((p[4] + p[5]) + (p[6] + p[7]));
  float e0 = hi ? -etot : etot;                                         // bit 8
  float e1 = (p[0] + p[1] + p[2] + p[3]) - (p[4] + p[5] + p[6] + p[7]); // bit 4
  float e2 = (p[0] + p[1]) - (p[2] + p[3]) + (p[4] + p[5]) - (p[6] + p[7]); // bit 2
  float e3 = (p[0] - p[1]) + (p[2] - p[3]) + (p[4] - p[5]) + (p[6] - p[7]); // bit 1

  // merge the two row-halves (lanes l and l^16 hold rows 0-7 / 8-15 of col n)
  e0 += __shfl_xor(e0, 16, 32);
  e1 += __shfl_xor(e1, 16, 32);
  e2 += __shfl_xor(e2, 16, 32);
  e3 += __shfl_xor(e3, 16, 32);

  if (!hi) {
    float4 o = {e0, e1, e2, e3};
    reinterpret_cast<float4*>(out)[s] = o;
  }
}

// ---------------------------------------------------------------------------
extern "C" void kernel_launch(void* const* d_in, const int* in_sizes, int n_in,
                              void* d_out, int out_size, void* d_ws,
                              size_t ws_size, hipStream_t stream) {
  const float* x = (const float*)d_in[0];        // [B,4] f32
  const float* w = (const float*)d_in[1];        // [4]   f32
  float* out = (float*)d_out;                    // [B,4] f32
  float* Ure = (float*)d_ws;                     // 256 f32
  float* Uim = Ure + 256;                        // 256 f32

  const int B = in_sizes[0] / 4;
  const int ntiles = B / 16;                     // B = 262144 -> 16384 tiles

  build_unitary_kernel<<<1, 16, 0, stream>>>(w, Ure, Uim);

  const int blocks = (ntiles + 3) / 4;           // 4 waves (tiles) per block
  quanv_wmma_kernel<<<blocks, 128, 0, stream>>>(x, Ure, Uim, out, ntiles);
}